// Attention_lora_46858093199373
// MI455X (gfx1250) — compile-verified
//
#include <hip/hip_runtime.h>
#include <hip/hip_bf16.h>
#include <math.h>

// ---------------- problem constants ----------------
#define BATCH   64
#define HH      14
#define WWID    14
#define PSEQ    196          // H*W
#define DIMC    768
#define NHEADS  12
#define HD      64
#define QKVDIM  2304         // 3*DIM
#define MTOT    (BATCH*PSEQ) // 12544
#define BHTOT   (BATCH*NHEADS) // 768
#define LORA_R  4

typedef float v2f __attribute__((ext_vector_type(2)));
typedef float v8f __attribute__((ext_vector_type(8)));

__device__ __forceinline__ v8f v8f_zero() {
    v8f z;
#pragma unroll
    for (int i = 0; i < 8; ++i) z[i] = 0.0f;
    return z;
}

// fp32 matrix core: D(16x16) = A(16x4) * B(4x16) + C   (V_WMMA_F32_16X16X4_F32)
__device__ __forceinline__ v8f wmma4(v2f a, v2f b, v8f c) {
    return __builtin_amdgcn_wmma_f32_16x16x4_f32(
        /*neg_a=*/false, a, /*neg_b=*/false, b,
        /*c_mod=*/(short)0, c, /*reuse_a=*/false, /*reuse_b=*/false);
}

// Async global->LDS copy, 16B per lane (GLOBAL_LOAD_ASYNC_TO_LDS_B128, GVS mode).
// INST_OFFSET is added to BOTH the LDS and global addresses (ISA 10.x), so one
// (ldsAddr, vOff) pair covers both 16B halves of a 32B slice via offset:0/16.
__device__ __forceinline__ void async_copy32B(unsigned ldsAddr, unsigned vOffBytes,
                                              const float* base) {
    asm volatile("global_load_async_to_lds_b128 %0, %1, %2 offset:0"
                 :: "v"(ldsAddr), "v"(vOffBytes), "s"(base) : "memory");
    asm volatile("global_load_async_to_lds_b128 %0, %1, %2 offset:16"
                 :: "v"(ldsAddr), "v"(vOffBytes), "s"(base) : "memory");
}

__device__ __forceinline__ void wait_asynccnt_le4() {
    asm volatile("s_wait_asynccnt 0x4" ::: "memory");
}
__device__ __forceinline__ void wait_asynccnt_0() {
    asm volatile("s_wait_asynccnt 0x0" ::: "memory");
}

// ---------------------------------------------------------------------------
// Kernel 1: fold LoRA rank-4 updates into effective QKV weight (2304 x 768)
// ---------------------------------------------------------------------------
__global__ void build_weff(const float* __restrict__ qkv_w,
                           const float* __restrict__ la_q, const float* __restrict__ lb_q,
                           const float* __restrict__ la_v, const float* __restrict__ lb_v,
                           float* __restrict__ weff) {
    int idx = blockIdx.x * blockDim.x + threadIdx.x;   // over 2304*768 exactly
    int n = idx / DIMC, k = idx % DIMC;
    float w = qkv_w[idx];
    if (n < DIMC) {
#pragma unroll
        for (int r = 0; r < LORA_R; ++r) w += lb_q[n * LORA_R + r] * la_q[r * DIMC + k];
    } else if (n >= 2 * DIMC) {
        int nv = n - 2 * DIMC;
#pragma unroll
        for (int r = 0; r < LORA_R; ++r) w += lb_v[nv * LORA_R + r] * la_v[r * DIMC + k];
    }
    weff[idx] = w;
}

// ---------------------------------------------------------------------------
// Kernel 2/6: C(MxN) = A(MxK) @ Wt(NxK)^T + bias.  M,N multiples of 64, K of 32.
// Block = 256 thr (8 waves). Block tile 64x64; wave tile 16x32 (2 accumulators).
// Global->LDS tiles move via ASYNC b128 copies with a double-buffered pipeline.
// ---------------------------------------------------------------------------
template <int KDIM>
__global__ __launch_bounds__(256) void gemm_wmma_bias(
    const float* __restrict__ A, const float* __restrict__ Wt,
    const float* __restrict__ bias, float* __restrict__ C, int Ncols) {
    __shared__ float As[2][64][36];  // 36-pad: 16B rows, conflict-free frag reads
    __shared__ float Bs[2][64][36];

    const int t    = threadIdx.x;
    const int lane = t & 31;
    const int wave = t >> 5;
    const int wm   = wave & 3;            // 0..3 : 16-row M slice
    const int wn   = wave >> 2;           // 0..1 : 32-col N slice
    const int l16  = lane & 15;
    const int koff = (lane < 16) ? 0 : 2;
    const int hi8  = (lane < 16) ? 0 : 8;
    const size_t m0 = (size_t)blockIdx.y * 64;
    const int    n0 = blockIdx.x * 64;

    const int ldRow = t >> 2;             // 0..63
    const int ldCol = (t & 3) * 8;        // 0,8,16,24  (32B slice per thread)

    // byte offsets (fit in 32 bits: A <= 116MB handled via 64-bit below if needed;
    // here A rows*K*4 <= ~115MB for qkv GEMM -> still < 2^32, ok)
    const unsigned aRow = (unsigned)(((size_t)(m0 + ldRow) * KDIM + ldCol) * 4u);
    const unsigned bRow = (unsigned)(((size_t)(n0 + ldRow) * KDIM + ldCol) * 4u);

    const unsigned ldsA0 = (unsigned)(size_t)&As[0][ldRow][ldCol];
    const unsigned ldsA1 = (unsigned)(size_t)&As[1][ldRow][ldCol];
    const unsigned ldsB0 = (unsigned)(size_t)&Bs[0][ldRow][ldCol];
    const unsigned ldsB1 = (unsigned)(size_t)&Bs[1][ldRow][ldCol];

    v8f acc0 = v8f_zero();
    v8f acc1 = v8f_zero();

    // prologue: stage 0 into buffer 0
    async_copy32B(ldsA0, aRow, A);
    async_copy32B(ldsB0, bRow, Wt);

    int buf = 0;
    for (int k0 = 0; k0 < KDIM; k0 += 32) {
        if (k0 + 32 < KDIM) {   // prefetch next stage into the other buffer
            unsigned kb = (unsigned)(k0 + 32) * 4u;
            async_copy32B(buf ? ldsA0 : ldsA1, aRow + kb, A);
            async_copy32B(buf ? ldsB0 : ldsB1, bRow + kb, Wt);
            wait_asynccnt_le4();     // current stage's 4 copies complete
        } else {
            wait_asynccnt_0();
        }
        __syncthreads();             // all waves' slices of `buf` are in LDS

        const float (*Ab)[36] = As[buf];
        const float (*Bb)[36] = Bs[buf];
#pragma unroll
        for (int kk = 0; kk < 32; kk += 4) {
            v2f a, b0, b1;
            a[0]  = Ab[wm * 16 + l16][kk + koff];
            a[1]  = Ab[wm * 16 + l16][kk + koff + 1];
            b0[0] = Bb[wn * 32 + l16][kk + koff];
            b0[1] = Bb[wn * 32 + l16][kk + koff + 1];
            b1[0] = Bb[wn * 32 + 16 + l16][kk + koff];
            b1[1] = Bb[wn * 32 + 16 + l16][kk + koff + 1];
            acc0 = wmma4(a, b0, acc0);
            acc1 = wmma4(a, b1, acc1);
        }
        __syncthreads();             // reads done before async refill of `buf`
        buf ^= 1;
    }

    const int nA = n0 + wn * 32 + l16;
    const int nB = nA + 16;
    const float biasA = bias ? bias[nA] : 0.0f;
    const float biasB = bias ? bias[nB] : 0.0f;
#pragma unroll
    for (int r = 0; r < 8; ++r) {
        size_t m = m0 + wm * 16 + r + hi8;
        C[m * Ncols + nA] = acc0[r] + biasA;
        C[m * Ncols + nB] = acc1[r] + biasB;
    }
}

// ---------------------------------------------------------------------------
// Kernel 3: per-head attention logits: scale*q@k^T + rel_h + rel_w
// grid (nTiles=4, mTiles=4, bh=768), block 256.  K = HD = 64 in one LDS stage.
// ---------------------------------------------------------------------------
__global__ __launch_bounds__(256) void attn_scores(
    const float* __restrict__ qkv, const float* __restrict__ rel_pos_h,
    const float* __restrict__ rel_pos_w, float* __restrict__ attn) {
    __shared__ float qs[64][68];
    __shared__ float ks[64][68];
    __shared__ float relh[64][14];
    __shared__ float relw[64][14];

    const int t    = threadIdx.x;
    const int bh   = blockIdx.z;
    const int bb   = bh / NHEADS;
    const int head = bh % NHEADS;
    const int m0   = blockIdx.y * 64;
    const int n0   = blockIdx.x * 64;

    {   // stage q/k head slices (row = token, 64 contiguous floats)
        int i  = t >> 2;
        int c0 = (t & 3) * 16;
        int mq = m0 + i; if (mq > PSEQ - 1) mq = PSEQ - 1;
        int nk = n0 + i; if (nk > PSEQ - 1) nk = PSEQ - 1;
        const float* qp = qkv + (size_t)(bb * PSEQ + mq) * QKVDIM + head * HD + c0;
        const float* kp = qkv + (size_t)(bb * PSEQ + nk) * QKVDIM + DIMC + head * HD + c0;
#pragma unroll
        for (int j = 0; j < 16; j += 4) {
            *(float4*)&qs[i][c0 + j] = *(const float4*)(qp + j);
            *(float4*)&ks[i][c0 + j] = *(const float4*)(kp + j);
        }
    }
    __syncthreads();

    // decomposed rel-pos dot products for the 64 query rows of this tile
    for (int idx = t; idx < 64 * 28; idx += 256) {
        int ml = idx / 28, c = idx % 28;
        int m = m0 + ml; if (m > PSEQ - 1) m = PSEQ - 1;
        int qi = m / WWID, qj = m % WWID;
        float s = 0.0f;
        if (c < 14) {
            const float* rp = rel_pos_h + (size_t)(qi - c + HH - 1) * HD;
#pragma unroll
            for (int d = 0; d < HD; ++d) s += qs[ml][d] * rp[d];
            relh[ml][c] = s;
        } else {
            int kw = c - 14;
            const float* rp = rel_pos_w + (size_t)(qj - kw + WWID - 1) * HD;
#pragma unroll
            for (int d = 0; d < HD; ++d) s += qs[ml][d] * rp[d];
            relw[ml][kw] = s;
        }
    }

    const int lane = t & 31;
    const int wave = t >> 5;
    const int wm = wave & 3, wn = wave >> 2;
    const int l16  = lane & 15;
    const int koff = (lane < 16) ? 0 : 2;
    const int hi8  = (lane < 16) ? 0 : 8;

    v8f acc0 = v8f_zero();
    v8f acc1 = v8f_zero();
#pragma unroll
    for (int kk = 0; kk < HD; kk += 4) {
        v2f a, b0, b1;
        a[0]  = qs[wm * 16 + l16][kk + koff];
        a[1]  = qs[wm * 16 + l16][kk + koff + 1];
        b0[0] = ks[wn * 32 + l16][kk + koff];
        b0[1] = ks[wn * 32 + l16][kk + koff + 1];
        b1[0] = ks[wn * 32 + 16 + l16][kk + koff];
        b1[1] = ks[wn * 32 + 16 + l16][kk + koff + 1];
        acc0 = wmma4(a, b0, acc0);
        acc1 = wmma4(a, b1, acc1);
    }
    __syncthreads();   // relh/relw fully written before reads below

    const float sc = 0.125f;   // 64^-0.5
    float* ab = attn + (size_t)bh * PSEQ * PSEQ;
#pragma unroll
    for (int r = 0; r < 8; ++r) {
        int ml = wm * 16 + r + hi8;
        int mg = m0 + ml;
        if (mg < PSEQ) {
            int ngA = n0 + wn * 32 + l16;
            int ngB = ngA + 16;
            if (ngA < PSEQ)
                ab[(size_t)mg * PSEQ + ngA] =
                    acc0[r] * sc + relh[ml][ngA / WWID] + relw[ml][ngA % WWID];
            if (ngB < PSEQ)
                ab[(size_t)mg * PSEQ + ngB] =
                    acc1[r] * sc + relh[ml][ngB / WWID] + relw[ml][ngB % WWID];
        }
    }
}

// ---------------------------------------------------------------------------
// Kernel 4: softmax over rows of length 196; one wave32 per row
// ---------------------------------------------------------------------------
__global__ __launch_bounds__(256) void softmax_rows(float* __restrict__ attn, int nrows) {
    int row  = blockIdx.x * 8 + (threadIdx.x >> 5);
    int lane = threadIdx.x & 31;
    if (row >= nrows) return;
    float* p = attn + (size_t)row * PSEQ;
    float vals[7];
    float mx = -3.0e38f;
#pragma unroll
    for (int j = 0; j < 7; ++j) {
        int c = lane + 32 * j;
        vals[j] = (c < PSEQ) ? p[c] : -3.0e38f;
        mx = fmaxf(mx, vals[j]);
    }
#pragma unroll
    for (int off = 16; off > 0; off >>= 1) mx = fmaxf(mx, __shfl_xor(mx, off, 32));
    float sum = 0.0f;
#pragma unroll
    for (int j = 0; j < 7; ++j) {
        int c = lane + 32 * j;
        float e = (c < PSEQ) ? __expf(vals[j] - mx) : 0.0f;
        vals[j] = e;
        sum += e;
    }
#pragma unroll
    for (int off = 16; off > 0; off >>= 1) sum += __shfl_xor(sum, off, 32);
    float inv = 1.0f / sum;
#pragma unroll
    for (int j = 0; j < 7; ++j) {
        int c = lane + 32 * j;
        if (c < PSEQ) p[c] = vals[j] * inv;
    }
}

// ---------------------------------------------------------------------------
// Kernel 5: out_pre = attn @ v  (per head: M=196, K=196, N=64)
// grid (mTiles=4, bh=768).  V tile transposed into LDS as Bs[d][n].
// ---------------------------------------------------------------------------
__global__ __launch_bounds__(256) void attn_v(
    const float* __restrict__ attn, const float* __restrict__ qkv,
    float* __restrict__ outp) {
    __shared__ float As[64][36];
    __shared__ float Bs[64][36];

    const int t    = threadIdx.x;
    const int bh   = blockIdx.y;
    const int bb   = bh / NHEADS;
    const int head = bh % NHEADS;
    const int m0   = blockIdx.x * 64;
    const float* Ab = attn + (size_t)bh * PSEQ * PSEQ;

    const int lane = t & 31;
    const int wave = t >> 5;
    const int wm = wave & 3, wn = wave >> 2;
    const int l16  = lane & 15;
    const int koff = (lane < 16) ? 0 : 2;
    const int hi8  = (lane < 16) ? 0 : 8;

    const int ldRow = t >> 2, ldCol = (t & 3) * 8;   // attn tile loader
    const int vRow  = t >> 3, vCol  = (t & 7) * 8;   // v tile loader (32 rows x 64 d)

    v8f acc0 = v8f_zero();
    v8f acc1 = v8f_zero();

    for (int k0 = 0; k0 < 224; k0 += 32) {           // 7 stages, last partial (zero-fill)
        float av[8], bv[8];
        int mA = m0 + ldRow;
#pragma unroll
        for (int j = 0; j < 8; ++j) {
            int kc = k0 + ldCol + j;
            av[j] = (mA < PSEQ && kc < PSEQ) ? Ab[(size_t)mA * PSEQ + kc] : 0.0f;
        }
        int nV = k0 + vRow;
        int nVc = (nV < PSEQ) ? nV : (PSEQ - 1);
        const float* vp = qkv + (size_t)(bb * PSEQ + nVc) * QKVDIM + 2 * DIMC + head * HD + vCol;
#pragma unroll
        for (int j = 0; j < 8; ++j) bv[j] = (nV < PSEQ) ? vp[j] : 0.0f;

        __syncthreads();
#pragma unroll
        for (int j = 0; j < 8; ++j) As[ldRow][ldCol + j] = av[j];
#pragma unroll
        for (int j = 0; j < 8; ++j) Bs[vCol + j][vRow] = bv[j];   // transpose: Bs[d][n]
        __syncthreads();
#pragma unroll
        for (int kk = 0; kk < 32; kk += 4) {
            v2f a, b0, b1;
            a[0]  = As[wm * 16 + l16][kk + koff];
            a[1]  = As[wm * 16 + l16][kk + koff + 1];
            b0[0] = Bs[wn * 32 + l16][kk + koff];
            b0[1] = Bs[wn * 32 + l16][kk + koff + 1];
            b1[0] = Bs[wn * 32 + 16 + l16][kk + koff];
            b1[1] = Bs[wn * 32 + 16 + l16][kk + koff + 1];
            acc0 = wmma4(a, b0, acc0);
            acc1 = wmma4(a, b1, acc1);
        }
    }

#pragma unroll
    for (int r = 0; r < 8; ++r) {
        int mg = m0 + wm * 16 + r + hi8;
        if (mg < PSEQ) {
            int d0 = wn * 32 + l16;
            float* op = outp + (size_t)(bb * PSEQ + mg) * DIMC + head * HD;
            op[d0]      = acc0[r];
            op[d0 + 16] = acc1[r];
        }
    }
}

// ---------------------------------------------------------------------------
extern "C" void kernel_launch(void* const* d_in, const int* in_sizes, int n_in,
                              void* d_out, int out_size, void* d_ws, size_t ws_size,
                              hipStream_t stream) {
    const float* x      = (const float*)d_in[0];
    const float* qkv_w  = (const float*)d_in[1];
    const float* qkv_b  = (const float*)d_in[2];
    const float* proj_w = (const float*)d_in[3];
    const float* proj_b = (const float*)d_in[4];
    const float* la_q   = (const float*)d_in[5];
    const float* lb_q   = (const float*)d_in[6];
    const float* la_v   = (const float*)d_in[7];
    const float* lb_v   = (const float*)d_in[8];
    const float* rel_h  = (const float*)d_in[9];
    const float* rel_w  = (const float*)d_in[10];
    float* out = (float*)d_out;

    // workspace layout (floats): weff | qkv | attn | out_pre  (~280 MB)
    float* weff = (float*)d_ws;
    float* qkv  = weff + (size_t)QKVDIM * DIMC;
    float* attn = qkv  + (size_t)MTOT * QKVDIM;
    float* outp = attn + (size_t)BHTOT * PSEQ * PSEQ;

    build_weff<<<(QKVDIM * DIMC) / 256, 256, 0, stream>>>(qkv_w, la_q, lb_q, la_v, lb_v, weff);
    gemm_wmma_bias<DIMC><<<dim3(QKVDIM / 64, MTOT / 64), 256, 0, stream>>>(
        x, weff, qkv_b, qkv, QKVDIM);
    attn_scores<<<dim3(4, 4, BHTOT), 256, 0, stream>>>(qkv, rel_h, rel_w, attn);
    softmax_rows<<<(BHTOT * PSEQ + 7) / 8, 256, 0, stream>>>(attn, BHTOT * PSEQ);
    attn_v<<<dim3(4, BHTOT), 256, 0, stream>>>(attn, qkv, outp);
    gemm_wmma_bias<DIMC><<<dim3(DIMC / 64, MTOT / 64), 256, 0, stream>>>(
        outp, proj_w, proj_b, out, DIMC);
}